// myHippo_27882927685769
// MI455X (gfx1250) — compile-verified
//
#include <hip/hip_runtime.h>
#include <math.h>

// Problem constants (from reference)
#define P_ROWS 65536
#define D_DIM  1024
#define TILES  (P_ROWS / 16)   // 4096 row-tiles of 16 rows
#define NBLK   512
#define TPB    256             // 8 wave32 waves
#define NWAVE  (TPB / 32)
#define EPSF   1e-8f

static_assert(TILES == NBLK * NWAVE, "one tile per wave");

typedef float v2f __attribute__((ext_vector_type(2)));
typedef float v8f __attribute__((ext_vector_type(8)));

// ds_swizzle SWAPX16: group-of-32, and=0x1f, or=0, xor=0x10 -> 0x401F
__device__ __forceinline__ float swz_swap16(float v) {
  return __int_as_float(__builtin_amdgcn_ds_swizzle(__float_as_int(v), 0x401F));
}

// D = A(16x4 f32) * B(4x16 f32) + C(16x16 f32), wave32 WMMA.
__device__ __forceinline__ v8f wmma_f32_16x16x4(v2f a, v2f b, v8f c) {
#if __has_builtin(__builtin_amdgcn_wmma_f32_16x16x4_f32)
  return __builtin_amdgcn_wmma_f32_16x16x4_f32(false, a, false, b, (short)0, c,
                                               false, false);
#else
  asm volatile("v_wmma_f32_16x16x4_f32 %0, %1, %2, %0"
               : "+v"(c)
               : "v"(a), "v"(b));
  return c;
#endif
}

__global__ void resonance_zero(float* __restrict__ out) {
  ((float4*)out)[threadIdx.x] = make_float4(0.f, 0.f, 0.f, 0.f);
}

// Fused single pass over mem:
//   per 16-row tile (one wave each): dots (WMMA) + row norms (VALU) -> sim,
//   then re-read tile (L2-hot) and accumulate sim_i * row_i into per-lane
//   output partials; block-reduce via LDS f32 atomics; global f32 atomics.
__global__ __launch_bounds__(TPB) void resonance_main(
    const float* __restrict__ x, const float* __restrict__ mem,
    float* __restrict__ out) {
  __shared__ float sx[D_DIM];          // x staged once per block
  __shared__ float sout[D_DIM];        // block output partial (also xnorm scratch)
  __shared__ float wscr[NWAVE * 32];   // per-wave: 16 dots + 16 sims
  __shared__ float sxinv;

  const int tid  = threadIdx.x;
  const int lane = tid & 31;
  const int wave = tid >> 5;
  const int half = lane >> 4;   // lane half: supplies WMMA K pairs
  const int m    = lane & 15;   // this lane's A-matrix row within the tile

  // Stage x into LDS + ||x||^2 reduction (deterministic, no FP atomics).
  {
    float4 xv = ((const float4*)x)[tid];
    ((float4*)sx)[tid] = xv;
    sout[tid] = xv.x * xv.x + xv.y * xv.y + xv.z * xv.z + xv.w * xv.w;
  }
  __syncthreads();
  if (tid == 0) {
    float s = 0.f;
    for (int i = 0; i < TPB; ++i) s += sout[i];
    sxinv = 1.0f / fmaxf(sqrtf(s), EPSF);
  }
  __syncthreads();
  const float xinv = sxinv;
  ((float4*)sout)[tid] = make_float4(0.f, 0.f, 0.f, 0.f);  // zero block partial
  __syncthreads();

  float4 oacc[8];
#pragma unroll
  for (int j = 0; j < 8; ++j) oacc[j] = make_float4(0.f, 0.f, 0.f, 0.f);

  float* wdot = wscr + wave * 32;
  float* wsim = wdot + 16;

  for (int t = blockIdx.x * NWAVE + wave; t < TILES; t += gridDim.x * NWAVE) {
    const size_t r0   = (size_t)t * 16;
    const float* rowp = mem + (r0 + (size_t)m) * D_DIM;  // lane's A row
    const float* ap   = rowp + 4 * half;                 // half h covers cols k0+4h..+3
    const float* bp   = sx + 4 * half;

    v8f   acc = {};
    float nrm = 0.f;
    // Documented A layout: (half h, vgpr v) holds K = 2h+v. B mirrors it, with
    // the x-chunk replicated into all 16 columns so every C column = dots.
    // One b128 load feeds two WMMAs: K-sets {k0,k0+1,k0+4,k0+5} and
    // {k0+2,k0+3,k0+6,k0+7}.
#pragma unroll 4
    for (int k0 = 0; k0 < D_DIM; k0 += 8) {
      if ((k0 & 24) == 0) __builtin_prefetch(ap + k0 + 512, 0, 0);
      float4 a4 = *(const float4*)(ap + k0);
      float4 b4 = *(const float4*)(bp + k0);
      v2f aA = {a4.x, a4.y};
      v2f bA = {b4.x, b4.y};
      v2f aB = {a4.z, a4.w};
      v2f bB = {b4.z, b4.w};
      acc = wmma_f32_16x16x4(aA, bA, acc);
      acc = wmma_f32_16x16x4(aB, bB, acc);
      nrm = fmaf(a4.x, a4.x, nrm);
      nrm = fmaf(a4.y, a4.y, nrm);
      nrm = fmaf(a4.z, a4.z, nrm);
      nrm = fmaf(a4.w, a4.w, nrm);
    }
    // Full ||row_m||^2: lanes l and l^16 hold complementary column halves.
    float nrmf = nrm + swz_swap16(nrm);
    float rinv = 1.0f / fmaxf(sqrtf(nrmf), EPSF);

    // All B columns equal -> lane half h holds dot(rows 8h..8h+7) in acc[0..7].
    if (m == 0) {
      *(float4*)(wdot + 8 * half)     = make_float4(acc[0], acc[1], acc[2], acc[3]);
      *(float4*)(wdot + 8 * half + 4) = make_float4(acc[4], acc[5], acc[6], acc[7]);
    }
    asm volatile("s_wait_dscnt 0" ::: "memory");   // wave-private LDS fence
    if (half == 0) wsim[m] = wdot[m] * rinv * xinv;
    asm volatile("s_wait_dscnt 0" ::: "memory");

    float4 sA = *(const float4*)(wsim + 0);
    float4 sB = *(const float4*)(wsim + 4);
    float4 sC = *(const float4*)(wsim + 8);
    float4 sD = *(const float4*)(wsim + 12);
    const float simr[16] = {sA.x, sA.y, sA.z, sA.w, sB.x, sB.y, sB.z, sB.w,
                            sC.x, sC.y, sC.z, sC.w, sD.x, sD.y, sD.z, sD.w};

    // out[d] += sim_i * mem[i][d]; tile re-read is coalesced (512B/inst) and L2-hot.
#pragma unroll
    for (int i = 0; i < 16; ++i) {
      const float* rp = mem + (r0 + (size_t)i) * D_DIM + lane * 4;
      const float  si = simr[i];
#pragma unroll
      for (int j = 0; j < 8; ++j) {
        float4 v = *(const float4*)(rp + j * 128);
        oacc[j].x = fmaf(si, v.x, oacc[j].x);
        oacc[j].y = fmaf(si, v.y, oacc[j].y);
        oacc[j].z = fmaf(si, v.z, oacc[j].z);
        oacc[j].w = fmaf(si, v.w, oacc[j].w);
      }
    }
  }

  // Wave partials -> block partial via LDS f32 atomics (ds_add_f32).
#pragma unroll
  for (int j = 0; j < 8; ++j) {
    int d = j * 128 + lane * 4;
    __hip_atomic_fetch_add(&sout[d + 0], oacc[j].x, __ATOMIC_RELAXED, __HIP_MEMORY_SCOPE_WORKGROUP);
    __hip_atomic_fetch_add(&sout[d + 1], oacc[j].y, __ATOMIC_RELAXED, __HIP_MEMORY_SCOPE_WORKGROUP);
    __hip_atomic_fetch_add(&sout[d + 2], oacc[j].z, __ATOMIC_RELAXED, __HIP_MEMORY_SCOPE_WORKGROUP);
    __hip_atomic_fetch_add(&sout[d + 3], oacc[j].w, __ATOMIC_RELAXED, __HIP_MEMORY_SCOPE_WORKGROUP);
  }
  __syncthreads();

  // Block partial -> global accumulator (global_atomic_add_f32).
#pragma unroll
  for (int e = 0; e < 4; ++e) {
    __hip_atomic_fetch_add(&out[tid * 4 + e], sout[tid * 4 + e],
                           __ATOMIC_RELAXED, __HIP_MEMORY_SCOPE_AGENT);
  }
}

// In-place: out /= max(|out|)
__global__ __launch_bounds__(256) void resonance_finalize(float* __restrict__ out) {
  __shared__ float red[256];
  __shared__ float sinv;
  const int tid = threadIdx.x;
  float4 v = ((float4*)out)[tid];
  red[tid] = fmaxf(fmaxf(fabsf(v.x), fabsf(v.y)), fmaxf(fabsf(v.z), fabsf(v.w)));
  __syncthreads();
  if (tid == 0) {
    float mx = 0.f;
    for (int i = 0; i < 256; ++i) mx = fmaxf(mx, red[i]);
    sinv = 1.0f / mx;
  }
  __syncthreads();
  const float inv = sinv;
  v.x *= inv; v.y *= inv; v.z *= inv; v.w *= inv;
  ((float4*)out)[tid] = v;
}

extern "C" void kernel_launch(void* const* d_in, const int* in_sizes, int n_in,
                              void* d_out, int out_size, void* d_ws, size_t ws_size,
                              hipStream_t stream) {
  (void)in_sizes; (void)n_in; (void)out_size; (void)d_ws; (void)ws_size;
  const float* x   = (const float*)d_in[0];   // [1024]
  const float* mem = (const float*)d_in[1];   // [65536, 1024]
  float*       out = (float*)d_out;           // [1024]

  resonance_zero<<<1, 256, 0, stream>>>(out);
  resonance_main<<<NBLK, TPB, 0, stream>>>(x, mem, out);
  resonance_finalize<<<1, 256, 0, stream>>>(out);
}